// FCGraphGNN_46729244180997
// MI455X (gfx1250) — compile-verified
//
#include <hip/hip_runtime.h>

// ---------------------------------------------------------------------------
// GCN: h1 = relu(Prop(x@W1)+b1); h2 = relu(Prop(h1@W2)+b2); out = pool@Wo+bo
// Prop(t)[d] = sum_e t[src]*dinv[src]*ew*dinv[dst]  + t[d]*dinv[d]^2 (selfloop)
// Dominant cost: 2 x (3.2M edges x 512B L2 traffic) ~ 3.3GB in L2 (h fits in
// 192MB L2). GEMM2 (50000x64x64 f32) uses v_wmma_f32_16x16x4_f32.
// ---------------------------------------------------------------------------

#define NN 50000
#define NE 3200000
#define NG 50
#define HID 64

typedef __attribute__((ext_vector_type(2))) float v2f;
typedef __attribute__((ext_vector_type(8))) float v8f;

// ---- stage 0: deg init (self-loop weight 1) + zero pool scratch -----------
__global__ void k_init(float* __restrict__ deg, float* __restrict__ pool, int nPool) {
    int i = blockIdx.x * blockDim.x + threadIdx.x;
    if (i < NN) deg[i] = 1.0f;
    if (i < nPool) pool[i] = 0.0f;
}

// ---- stage 1: weighted degree over edges ----------------------------------
__global__ void k_deg(const int* __restrict__ dst, const float* __restrict__ ew,
                      float* __restrict__ deg) {
    int e = blockIdx.x * blockDim.x + threadIdx.x;
    if (e < NE) atomicAdd(&deg[dst[e]], ew[e]);
}

// ---- stage 2: dinv = rsqrt(deg) in place ----------------------------------
__global__ void k_dinv(float* __restrict__ deg) {
    int i = blockIdx.x * blockDim.x + threadIdx.x;
    if (i < NN) {
        float d = deg[i];
        deg[i] = (d > 0.0f) ? rsqrtf(d) : 0.0f;
    }
}

// ---- GEMM1: T = x @ W1   (Nx5 @ 5x64), K=5 -> scalar, W1 in LDS -----------
__global__ void k_gemm1(const float* __restrict__ x, const float* __restrict__ W1,
                        float* __restrict__ T) {
    __shared__ float sW[5 * HID];
    for (int i = threadIdx.x; i < 5 * HID; i += blockDim.x) sW[i] = W1[i];
    __syncthreads();
    int gid = blockIdx.x * blockDim.x + threadIdx.x;  // 4 nodes / 256-thread block
    int node = gid >> 6, j = gid & 63;
    if (node >= NN) return;
    const float* xr = x + (size_t)node * 5;
    float acc = 0.0f;
#pragma unroll
    for (int k = 0; k < 5; ++k) acc = fmaf(xr[k], sW[k * HID + j], acc);
    T[(size_t)node * HID + j] = acc;
}

// ---- zero an N*64 f32 buffer ----------------------------------------------
__global__ void k_zero(float* __restrict__ p, int n) {
    int i = blockIdx.x * blockDim.x + threadIdx.x;
    if (i < n) p[i] = 0.0f;
}

// ---- edge scatter: agg[dst] += T[src] * dinv[src]*ew*dinv[dst] ------------
__global__ void k_scatter(const float* __restrict__ T, const int* __restrict__ src,
                          const int* __restrict__ dst, const float* __restrict__ ew,
                          const float* __restrict__ dinv, float* __restrict__ agg) {
    long long gid = (long long)blockIdx.x * blockDim.x + threadIdx.x;
    int e = (int)(gid >> 6), j = (int)(gid & 63);
    if (e >= NE) return;
    int s = src[e], d = dst[e];
    float nrm = dinv[s] * ew[e] * dinv[d];
    atomicAdd(&agg[(size_t)d * HID + j], T[(size_t)s * HID + j] * nrm);
}

// ---- self-loop + bias + relu: h = relu(agg + T*dinv^2 + b) (in place) -----
__global__ void k_bias_relu(const float* __restrict__ T, const float* __restrict__ dinv,
                            const float* __restrict__ b, float* __restrict__ agg) {
    int gid = blockIdx.x * blockDim.x + threadIdx.x;
    int node = gid >> 6, j = gid & 63;
    if (node >= NN) return;
    float di = dinv[node];
    size_t o = (size_t)node * HID + j;
    float v = agg[o] + T[o] * di * di + b[j];
    agg[o] = v > 0.0f ? v : 0.0f;
}

// ---- GEMM2 (WMMA): T = H @ W2  (Nx64 @ 64x64), f32 16x16x4 ---------------
// Block = 128 threads = 4 waves; wave w owns the N-tile n0 = 16*w.
// Block b owns M-tile rows [16b, 16b+16).  50000 = 3125*16 exactly -> EXEC
// stays all-ones (WMMA requirement), no tail handling needed.
__global__ void k_gemm2_wmma(const float* __restrict__ H, const float* __restrict__ W,
                             float* __restrict__ T) {
    const int wave = threadIdx.x >> 5;
    const int lane = threadIdx.x & 31;
    const int lhalf = lane >> 4;    // 0: lanes 0-15, 1: lanes 16-31
    const int l16  = lane & 15;
    const int n0   = wave * 16;
    const int m0   = blockIdx.x * 16;

    // B fragments for all 16 k-steps (W2 = 16KB, L1/L2 resident).
    // ISA B 4x16 layout: VGPR v holds row K=v (lanes 0-15) / K=v+2 (lanes 16-31).
    v2f bfrag[16];
#pragma unroll
    for (int s = 0; s < 16; ++s) {
        int k = 4 * s + 2 * lhalf;
        bfrag[s].x = W[(size_t)k * HID + n0 + l16];
        bfrag[s].y = W[(size_t)(k + 1) * HID + n0 + l16];
    }

    v8f acc = {};
    const float* arow = H + (size_t)(m0 + l16) * HID;
#pragma unroll
    for (int s = 0; s < 16; ++s) {
        // ISA A 16x4 layout: lane<16 holds K={k,k+1} of row M=lane;
        // lane>=16 holds K={k+2,k+3} of row M=lane-16.
        int k = 4 * s + 2 * lhalf;
        v2f a;
        a.x = arow[k];
        a.y = arow[k + 1];
        acc = __builtin_amdgcn_wmma_f32_16x16x4_f32(
            /*neg_a=*/false, a, /*neg_b=*/false, bfrag[s],
            /*c_mod=*/(short)0, acc, /*reuse_a=*/false, /*reuse_b=*/false);
    }

    // C/D 16x16 f32 layout: VGPR r -> D[M=r + 8*lhalf][N=l16].
#pragma unroll
    for (int r = 0; r < 8; ++r)
        T[(size_t)(m0 + r + 8 * lhalf) * HID + n0 + l16] = acc[r];
}

// ---- global mean pool: sums[g] += h[node], cnt[g] += 1 --------------------
__global__ void k_pool(const float* __restrict__ H, const int* __restrict__ batch,
                       float* __restrict__ sums, float* __restrict__ cnt) {
    int gid = blockIdx.x * blockDim.x + threadIdx.x;
    int node = gid >> 6, j = gid & 63;
    if (node >= NN) return;
    int g = batch[node];
    atomicAdd(&sums[(size_t)g * HID + j], H[(size_t)node * HID + j]);
    if (j == 0) atomicAdd(&cnt[g], 1.0f);
}

// ---- head: out[g][o] = (sums[g]/max(cnt,1)) . Wo[:,o] + bo[o] -------------
__global__ void k_out(const float* __restrict__ sums, const float* __restrict__ cnt,
                      const float* __restrict__ Wo, const float* __restrict__ bo,
                      float* __restrict__ out) {
    int t = threadIdx.x;
    if (t >= NG * 2) return;
    int g = t >> 1, o = t & 1;
    float inv = 1.0f / fmaxf(cnt[g], 1.0f);
    float acc = bo[o];
#pragma unroll
    for (int j = 0; j < HID; ++j)
        acc = fmaf(sums[(size_t)g * HID + j] * inv, Wo[j * 2 + o], acc);
    out[g * 2 + o] = acc;
}

extern "C" void kernel_launch(void* const* d_in, const int* in_sizes, int n_in,
                              void* d_out, int out_size, void* d_ws, size_t ws_size,
                              hipStream_t stream) {
    const float* x    = (const float*)d_in[0];
    const int*   ei   = (const int*)d_in[1];   // [2, NE]: row0=src, row1=dst
    const float* ew   = (const float*)d_in[2]; // [NE,1]
    const int*   batch= (const int*)d_in[3];
    const float* W1   = (const float*)d_in[4];
    const float* b1   = (const float*)d_in[5];
    const float* W2   = (const float*)d_in[6];
    const float* b2   = (const float*)d_in[7];
    const float* Wo   = (const float*)d_in[8];
    const float* bo   = (const float*)d_in[9];
    float* out = (float*)d_out;

    const int* src = ei;
    const int* dst = ei + NE;

    // workspace layout (floats): dinv[NN] | bufA[NN*64] | bufB[NN*64] | sums[NG*64] | cnt[NG]
    float* dinv = (float*)d_ws;
    float* bufA = dinv + NN;
    float* bufB = bufA + (size_t)NN * HID;
    float* sums = bufB + (size_t)NN * HID;
    float* cnt  = sums + (size_t)NG * HID;
    const int nPool = NG * HID + NG;

    const int nodeBlocks = (NN + 255) / 256;            // 196
    const int edgeBlocks = (NE + 255) / 256;            // 12500
    const int nhBlocks   = (NN * HID + 255) / 256;      // 12500 (node*64 grid)
    const long long scatterThreads = (long long)NE * HID;
    const int scatterBlocks = (int)((scatterThreads + 255) / 256); // 800000
    const int mTiles = NN / 16;                         // 3125 (exact)

    // normalization
    k_init<<<nodeBlocks, 256, 0, stream>>>(dinv, sums, nPool);
    k_deg<<<edgeBlocks, 256, 0, stream>>>(dst, ew, dinv);
    k_dinv<<<nodeBlocks, 256, 0, stream>>>(dinv);

    // layer 1
    k_gemm1<<<nhBlocks, 256, 0, stream>>>(x, W1, bufA);
    k_zero<<<nhBlocks, 256, 0, stream>>>(bufB, NN * HID);
    k_scatter<<<scatterBlocks, 256, 0, stream>>>(bufA, src, dst, ew, dinv, bufB);
    k_bias_relu<<<nhBlocks, 256, 0, stream>>>(bufA, dinv, b1, bufB);   // h1 in bufB

    // layer 2 (WMMA GEMM)
    k_gemm2_wmma<<<mTiles, 128, 0, stream>>>(bufB, W2, bufA);          // t2 in bufA
    k_zero<<<nhBlocks, 256, 0, stream>>>(bufB, NN * HID);
    k_scatter<<<scatterBlocks, 256, 0, stream>>>(bufA, src, dst, ew, dinv, bufB);
    k_bias_relu<<<nhBlocks, 256, 0, stream>>>(bufA, dinv, b2, bufB);   // h2 in bufB

    // pool + head
    k_pool<<<nhBlocks, 256, 0, stream>>>(bufB, batch, sums, cnt);
    k_out<<<1, 128, 0, stream>>>(sums, cnt, Wo, bo, out);
}